// FullFEMSolver_23673859736318
// MI455X (gfx1250) — compile-verified
//
#include <hip/hip_runtime.h>

// ---------------------------------------------------------------------------
// Batched diagonal-Hessian Newton-CG solver for MI455X (gfx1250, wave32).
// All vector state lives in d_ws (~50 MB -> resident in 192 MB L2).
// Reductions: per-block partials + deterministic fixed-order re-reduce by
// consumers. Wave-level sums use V_WMMA_F32_16X16X4_F32 (ones-B trick):
// one WMMA reduces 64 f32 (2 VGPRs x 32 lanes) to 16 row sums.
// ---------------------------------------------------------------------------

#define TPB 256
#define EPT 4
#define CHUNK (TPB * EPT)                      // 1024 elements per block
#define B_ 8
#define DOF_ 300000
#define CPB ((DOF_ + CHUNK - 1) / CHUNK)       // 293 blocks per batch
#define NEWTON_ITERS_ 6
#define CG_ITERS_ 20
#define BETA4 0.4f                              // 4*BETA
#define BETA12 1.2f                             // 12*BETA
#define ALPHA_MIN_ 0.05f

typedef float v2f __attribute__((ext_vector_type(2)));
typedef float v8f __attribute__((ext_vector_type(8)));

__constant__ float c_alphas[8] = {1.0f, 0.5f, 0.25f, 0.125f, 0.0625f, 0.05f, 0.05f, 0.05f};

// ---- wave-wide sum of (a0 + a1) over all 32 lanes via WMMA f32 16x16x4 ----
// A = per-lane data (2 VGPRs = the full 16x4 matrix), B = ones, C = 0.
// D[m][n] = sum_k A[m][k]  (row sums). Per-lane sum of the 8 D VGPRs gives
// rows 0..7 (lanes 0-15) / rows 8..15 (lanes 16-31); xor-16 combines halves.
__device__ inline float wave_sum64(float a0, float a1) {
  v2f A; A[0] = a0; A[1] = a1;
  v2f Bn; Bn[0] = 1.0f; Bn[1] = 1.0f;
  v8f C = {};
  v8f D = __builtin_amdgcn_wmma_f32_16x16x4_f32(false, A, false, Bn,
                                                (short)0, C, false, false);
  float s = D[0] + D[1] + D[2] + D[3] + D[4] + D[5] + D[6] + D[7];
  s += __shfl_xor(s, 16, 32);
  return s;
}

__device__ inline float block_sum(float a0, float a1, float* lds8) {
  float s = wave_sum64(a0, a1);
  int wid = threadIdx.x >> 5;
  __syncthreads();                      // protect lds8 reuse across calls
  if ((threadIdx.x & 31) == 0) lds8[wid] = s;
  __syncthreads();
  float tot = 0.0f;
#pragma unroll
  for (int w = 0; w < TPB / 32; ++w) tot += lds8[w];
  return tot;                           // same value in every thread
}

// Deterministic re-reduction of one batch's CPB per-block partials.
__device__ inline float reduce_partials(const float* __restrict__ p, float* lds8) {
  float a0 = 0.0f;
  for (int c = threadIdx.x; c < CPB; c += TPB) a0 += p[c];
  return block_sum(a0, 0.0f, lds8);
}

// Backtracking line-search alpha selection (S[0] = ||initial residual||^2).
__device__ inline float select_alpha(const float* __restrict__ LS, int b, float* lds8) {
  float S[9];
#pragma unroll
  for (int t = 0; t < 9; ++t)
    S[t] = reduce_partials(LS + (size_t)(b * 9 + t) * CPB, lds8);
  float a = ALPHA_MIN_;
  bool found = false;
#pragma unroll
  for (int t = 1; t < 9; ++t) {
    if (!found && S[t] < S[0]) { a = c_alphas[t - 1]; found = true; }
  }
  return a;
}

// ---------------------------------------------------------------------------
__global__ __launch_bounds__(TPB) void k_mask_init(float* __restrict__ mask) {
  int d = blockIdx.x * TPB + threadIdx.x;
  if (d < DOF_) mask[d] = 1.0f;
}

__global__ __launch_bounds__(TPB) void k_mask_scatter(float* __restrict__ mask,
                                                      const int* __restrict__ fx, int n) {
  int i = blockIdx.x * TPB + threadIdx.x;
  if (i < n) {
    int d = fx[i];
    if (d >= 0 && d < DOF_) mask[d] = 0.0f;
  }
}

// Fused: u_{n} = u_{n-1} + alpha_sel * (mask*x)   [if applyUpdate]
//        r = p = -(mask*(f - grad(u))), x = 0, partial sum of r^2 -> RR0
__global__ __launch_bounds__(TPB) void k_step(
    const float* __restrict__ usrc, float* __restrict__ U,
    const float* __restrict__ f, const float* __restrict__ kd,
    const float* __restrict__ mask, float* __restrict__ X,
    float* __restrict__ R, float* __restrict__ P,
    float* __restrict__ RR0, const float* __restrict__ LS, int applyUpdate) {
  __shared__ float lds8[TPB / 32];
  int b = blockIdx.x / CPB, c = blockIdx.x % CPB;
  float asel = 0.0f;
  if (applyUpdate) asel = select_alpha(LS, b, lds8);
  float a0 = 0.0f, a1 = 0.0f;
#pragma unroll
  for (int kq = 0; kq < EPT; ++kq) {
    int d = c * CHUNK + kq * TPB + threadIdx.x;
    if (d < DOF_) {
      int i = b * DOF_ + d;
      float m = mask[d];
      float u = usrc[i];
      if (applyUpdate) u += asel * (m * X[i]);
      U[i] = u;
      float g = kd[d] * u + BETA4 * u * u * u;
      float r = -(m * (f[i] - g));          // b = -filtered; r0 = b
      R[i] = r; P[i] = r; X[i] = 0.0f;
      float q = r * r;                       // == filtered^2
      if (kq < 2) a0 += q; else a1 += q;
    }
  }
  float tot = block_sum(a0, a1, lds8);
  if (threadIdx.x == 0) RR0[b * CPB + c] = tot;
}

// CG phase A: beta = RNS[j]/RNS[j-1] (guarded), p = r + beta*p,
//             Ap = mask * (k + 12*BETA*u^2) * p, partial p.Ap -> PP
__global__ __launch_bounds__(TPB) void k_cg_ap(
    const float* __restrict__ U, const float* __restrict__ kd,
    const float* __restrict__ mask, const float* __restrict__ R,
    float* __restrict__ P, float* __restrict__ AP,
    const float* __restrict__ RRcur, const float* __restrict__ RRprev,
    float* __restrict__ PP, int j) {
  __shared__ float lds8[TPB / 32];
  int b = blockIdx.x / CPB, c = blockIdx.x % CPB;
  float beta = 0.0f;
  if (j > 0) {
    float rc = reduce_partials(RRcur + b * CPB, lds8);
    float rp = reduce_partials(RRprev + b * CPB, lds8);
    beta = (rp > 0.0f) ? rc / rp : 0.0f;
  }
  float a0 = 0.0f, a1 = 0.0f;
#pragma unroll
  for (int kq = 0; kq < EPT; ++kq) {
    int d = c * CHUNK + kq * TPB + threadIdx.x;
    if (d < DOF_) {
      int i = b * DOF_ + d;
      float p = P[i];
      if (j > 0) { p = R[i] + beta * p; P[i] = p; }
      float u = U[i];
      float Ap = mask[d] * (kd[d] + BETA12 * u * u) * p;
      AP[i] = Ap;
      float q = p * Ap;
      if (kq < 2) a0 += q; else a1 += q;
    }
  }
  float tot = block_sum(a0, a1, lds8);
  if (threadIdx.x == 0) PP[b * CPB + c] = tot;
}

// CG phase B: alpha = rns/pAp (guarded), x += alpha*p, r -= alpha*Ap,
//             partial r.r -> RRnext (ping-pong buffer)
__global__ __launch_bounds__(TPB) void k_cg_upd(
    const float* __restrict__ P, const float* __restrict__ AP,
    float* __restrict__ X, float* __restrict__ R,
    const float* __restrict__ RRcur, const float* __restrict__ PP,
    float* __restrict__ RRnext) {
  __shared__ float lds8[TPB / 32];
  int b = blockIdx.x / CPB, c = blockIdx.x % CPB;
  float rns = reduce_partials(RRcur + b * CPB, lds8);
  float pap = reduce_partials(PP + b * CPB, lds8);
  float alpha = (fabsf(pap) > 0.0f) ? rns / pap : 0.0f;
  float a0 = 0.0f, a1 = 0.0f;
#pragma unroll
  for (int kq = 0; kq < EPT; ++kq) {
    int d = c * CHUNK + kq * TPB + threadIdx.x;
    if (d < DOF_) {
      int i = b * DOF_ + d;
      float x = X[i] + alpha * P[i];  X[i] = x;
      float r = R[i] - alpha * AP[i]; R[i] = r;
      float q = r * r;
      if (kq < 2) a0 += q; else a1 += q;
    }
  }
  float tot = block_sum(a0, a1, lds8);
  if (threadIdx.x == 0) RRnext[b * CPB + c] = tot;
}

// Line search: 9 candidates (t=0 is alpha=0 -> initial residual norm).
__global__ __launch_bounds__(TPB) void k_ls(
    const float* __restrict__ U, const float* __restrict__ X,
    const float* __restrict__ f, const float* __restrict__ kd,
    const float* __restrict__ mask, float* __restrict__ LS) {
  __shared__ float lds8[TPB / 32];
  int b = blockIdx.x / CPB, c = blockIdx.x % CPB;
  float acc0[9], acc1[9];
#pragma unroll
  for (int t = 0; t < 9; ++t) { acc0[t] = 0.0f; acc1[t] = 0.0f; }
#pragma unroll
  for (int kq = 0; kq < EPT; ++kq) {
    int d = c * CHUNK + kq * TPB + threadIdx.x;
    if (d < DOF_) {
      int i = b * DOF_ + d;
      float m = mask[d];
      float u = U[i];
      float du = m * X[i];
      float fi = f[i];
      float kk = kd[d];
#pragma unroll
      for (int t = 0; t < 9; ++t) {
        float at = (t == 0) ? 0.0f : c_alphas[t - 1];
        float cd = u + at * du;
        float cr = m * (fi - (kk * cd + BETA4 * cd * cd * cd));
        float q = cr * cr;
        if (kq < 2) acc0[t] += q; else acc1[t] += q;
      }
    }
  }
#pragma unroll
  for (int t = 0; t < 9; ++t) {
    float tot = block_sum(acc0[t], acc1[t], lds8);
    if (threadIdx.x == 0) LS[(size_t)(b * 9 + t) * CPB + c] = tot;
  }
}

// Final Newton update -> d_out
__global__ __launch_bounds__(TPB) void k_fin(
    const float* __restrict__ U, const float* __restrict__ X,
    const float* __restrict__ mask, const float* __restrict__ LS,
    float* __restrict__ out) {
  __shared__ float lds8[TPB / 32];
  int b = blockIdx.x / CPB, c = blockIdx.x % CPB;
  float asel = select_alpha(LS, b, lds8);
#pragma unroll
  for (int kq = 0; kq < EPT; ++kq) {
    int d = c * CHUNK + kq * TPB + threadIdx.x;
    if (d < DOF_) {
      int i = b * DOF_ + d;
      out[i] = U[i] + asel * (mask[d] * X[i]);
    }
  }
}

// ---------------------------------------------------------------------------
extern "C" void kernel_launch(void* const* d_in, const int* in_sizes, int n_in,
                              void* d_out, int out_size, void* d_ws, size_t ws_size,
                              hipStream_t stream) {
  const float* f  = (const float*)d_in[0];
  const float* u0 = (const float*)d_in[1];
  const float* kd = (const float*)d_in[2];
  const int*   fx = (const int*)d_in[3];
  int nfx = in_sizes[3];

  // Workspace layout (256B aligned)
  size_t off = 0;
  auto alloc = [&](size_t nfloats) {
    float* p = (float*)((char*)d_ws + off);
    off += ((nfloats * sizeof(float) + 255) / 256) * 256;
    return p;
  };
  float* MASK = alloc(DOF_);
  float* U    = alloc((size_t)B_ * DOF_);
  float* X    = alloc((size_t)B_ * DOF_);
  float* R    = alloc((size_t)B_ * DOF_);
  float* P    = alloc((size_t)B_ * DOF_);
  float* AP   = alloc((size_t)B_ * DOF_);
  float* RRa  = alloc((size_t)B_ * CPB);
  float* RRb  = alloc((size_t)B_ * CPB);
  float* PP   = alloc((size_t)B_ * CPB);
  float* LS   = alloc((size_t)B_ * 9 * CPB);
  float* RR[2] = {RRa, RRb};

  dim3 grid(B_ * CPB), blk(TPB);

  k_mask_init<<<(DOF_ + TPB - 1) / TPB, TPB, 0, stream>>>(MASK);
  k_mask_scatter<<<(nfx + TPB - 1) / TPB, TPB, 0, stream>>>(MASK, fx, nfx);

  for (int n = 0; n < NEWTON_ITERS_; ++n) {
    k_step<<<grid, blk, 0, stream>>>(n == 0 ? u0 : U, U, f, kd, MASK,
                                     X, R, P, RR[0], LS, n > 0 ? 1 : 0);
    for (int j = 0; j < CG_ITERS_; ++j) {
      k_cg_ap<<<grid, blk, 0, stream>>>(U, kd, MASK, R, P, AP,
                                        RR[j & 1], RR[(j + 1) & 1], PP, j);
      k_cg_upd<<<grid, blk, 0, stream>>>(P, AP, X, R,
                                         RR[j & 1], PP, RR[(j + 1) & 1]);
    }
    k_ls<<<grid, blk, 0, stream>>>(U, X, f, kd, MASK, LS);
  }
  k_fin<<<grid, blk, 0, stream>>>(U, X, MASK, LS, (float*)d_out);
}